// GATConv_42889543418170
// MI455X (gfx1250) — compile-verified
//
#include <hip/hip_runtime.h>

#define Bsz   2
#define INF   32
#define Hh    4
#define OUTo  16
#define HO    64           // H*OUT
#define BH    (Bsz*Hh)     // 8
#define NEG_SLOPE 0.2f

typedef float v2f __attribute__((ext_vector_type(2)));
typedef float v8f __attribute__((ext_vector_type(8)));

// ---------- order-preserving float<->uint key (for atomicMax segment-max) ----
__device__ __forceinline__ unsigned f2key(float f) {
    unsigned u = __float_as_uint(f);
    return (u & 0x80000000u) ? ~u : (u | 0x80000000u);
}
__device__ __forceinline__ float key2f(unsigned k) {
    unsigned u = (k & 0x80000000u) ? (k & 0x7FFFFFFFu) : ~k;
    return __uint_as_float(u);
}

// ---------------------------------------------------------------------------
// Kernel 1: ft = [feat[:taxi]@W1 ; feat[taxi:]@W2]   via V_WMMA_F32_16X16X4_F32
// One wave per 16-row tile; 4 column tiles (N=64); K=32 as 8 chained x4 steps.
// ---------------------------------------------------------------------------
__global__ __launch_bounds__(128) void gemm_ft_kernel(
    const float* __restrict__ feat, const float* __restrict__ W1,
    const float* __restrict__ W2, const int* __restrict__ taxi_ptr,
    float* __restrict__ ft, int Mtotal)
{
    const int wave = threadIdx.x >> 5;
    const int lane = threadIdx.x & 31;
    const int tile = blockIdx.x * 4 + wave;
    const int row_base = tile * 16;
    if (row_base >= Mtotal) return;              // wave-uniform guard

    const int taxi = *taxi_ptr;
    const float* __restrict__ W = ((row_base / Bsz) < taxi) ? W1 : W2;

    const int mrow = lane & 15;                  // row within tile (A) / col (B,D)
    const int koff = (lane >> 4) << 1;           // 0 or 2 (K split across half-waves)
    const float* __restrict__ arow = feat + (size_t)(row_base + mrow) * INF;

    v8f acc0 = {}, acc1 = {}, acc2 = {}, acc3 = {};

#pragma unroll
    for (int k0 = 0; k0 < INF; k0 += 4) {
        v2f a;
        a.x = arow[k0 + koff];
        a.y = arow[k0 + koff + 1];

        const float* __restrict__ bp = W + (size_t)(k0 + koff) * HO + mrow;
        v2f b0, b1, b2, b3;
        b0.x = bp[0];       b0.y = bp[HO + 0];
        b1.x = bp[16];      b1.y = bp[HO + 16];
        b2.x = bp[32];      b2.y = bp[HO + 32];
        b3.x = bp[48];      b3.y = bp[HO + 48];

        acc0 = __builtin_amdgcn_wmma_f32_16x16x4_f32(false, a, false, b0, (short)0, acc0, false, false);
        acc1 = __builtin_amdgcn_wmma_f32_16x16x4_f32(false, a, false, b1, (short)0, acc1, false, false);
        acc2 = __builtin_amdgcn_wmma_f32_16x16x4_f32(false, a, false, b2, (short)0, acc2, false, false);
        acc3 = __builtin_amdgcn_wmma_f32_16x16x4_f32(false, a, false, b3, (short)0, acc3, false, false);
    }

    // D layout: lane L holds column (L&15); VGPR r -> row r + 8*(L>=16)
    const int rowoff = (lane >> 4) << 3;         // 0 or 8
    float* __restrict__ out = ft + (size_t)row_base * HO + mrow;
#pragma unroll
    for (int r = 0; r < 8; ++r) {
        const size_t ro = (size_t)(r + rowoff) * HO;
        out[ro + 0]  = acc0[r];
        out[ro + 16] = acc1[r];
        out[ro + 32] = acc2[r];
        out[ro + 48] = acc3[r];
    }
}

// ---------------------------------------------------------------------------
// Kernel 2: el[n,b,h] = sum_o ft[n,b,h,o] * attn_l[h,o]     (er == el)
// ---------------------------------------------------------------------------
__global__ void el_kernel(const float* __restrict__ ft,
                          const float* __restrict__ attn_l,
                          float* __restrict__ el, int total /* N*B*H */)
{
    int idx = blockIdx.x * blockDim.x + threadIdx.x;
    if (idx >= total) return;
    const int h   = idx & (Hh - 1);
    const int row = idx >> 2;                    // n*B + b
    const float* f  = ft + (size_t)row * HO + h * OUTo;
    const float* al = attn_l + h * OUTo;
    float s = 0.f;
#pragma unroll
    for (int o = 0; o < OUTo; ++o) s += f[o] * al[o];
    el[idx] = s;
}

// ---------------------------------------------------------------------------
// Kernel 3: e = leaky_relu(el[src] + el[dst]); atomic segment-max into mkey[dst]
// ---------------------------------------------------------------------------
__global__ void edge_max_kernel(const float* __restrict__ el,
                                const int* __restrict__ src,
                                const int* __restrict__ dst,
                                float* __restrict__ ebuf,
                                unsigned* __restrict__ mkey, int total /* E*B*H */)
{
    int idx = blockIdx.x * blockDim.x + threadIdx.x;
    if (idx >= total) return;
    const int bh = idx & (BH - 1);
    const int e  = idx >> 3;
    const int s  = src[e];
    const int d  = dst[e];
    float v = el[s * BH + bh] + el[d * BH + bh];
    v = (v >= 0.f) ? v : NEG_SLOPE * v;
    ebuf[idx] = v;
    atomicMax(&mkey[d * BH + bh], f2key(v));
}

// ---------------------------------------------------------------------------
// Kernel 4: ex = exp(e - m[dst]); atomic segment-sum into ssum[dst]
// (ebuf overwritten in place: each thread reads then writes its own slot)
// ---------------------------------------------------------------------------
__global__ void edge_exp_kernel(float* ebuf,
                                const unsigned* __restrict__ mkey,
                                const int* __restrict__ dst,
                                float* __restrict__ ssum, int total /* E*B*H */)
{
    int idx = blockIdx.x * blockDim.x + threadIdx.x;
    if (idx >= total) return;
    const int bh = idx & (BH - 1);
    const int e  = idx >> 3;
    const int d  = dst[e];
    const float m  = key2f(mkey[d * BH + bh]);
    const float ex = expf(ebuf[idx] - m);
    ebuf[idx] = ex;
    atomicAdd(&ssum[d * BH + bh], ex);
}

// ---------------------------------------------------------------------------
// Kernel 5: a = ex / s[dst]   (second output)
// ---------------------------------------------------------------------------
__global__ void edge_norm_kernel(const float* __restrict__ exbuf,
                                 const float* __restrict__ ssum,
                                 const int* __restrict__ dst,
                                 float* __restrict__ a_out, int total /* E*B*H */)
{
    int idx = blockIdx.x * blockDim.x + threadIdx.x;
    if (idx >= total) return;
    const int bh = idx & (BH - 1);
    const int e  = idx >> 3;
    a_out[idx] = exbuf[idx] / ssum[dst[e] * BH + bh];
}

// ---------------------------------------------------------------------------
// Kernel 6: rst[dst] += ft[src] * a      (scatter, L2-resident atomics)
// ---------------------------------------------------------------------------
__global__ void aggregate_kernel(const float* __restrict__ ft,
                                 const float* __restrict__ a_out,
                                 const int* __restrict__ src,
                                 const int* __restrict__ dst,
                                 float* rst, int total /* E*B*H*OUT */)
{
    int idx = blockIdx.x * blockDim.x + threadIdx.x;
    if (idx >= total) return;
    const int o  = idx & (OUTo - 1);
    const int g  = idx >> 4;                     // e*B*H + bh
    const int bh = g & (BH - 1);
    const int e  = g >> 3;
    const int b  = bh >> 2;
    const int h  = bh & (Hh - 1);
    const int s  = src[e];
    const int d  = dst[e];
    const float a = a_out[g];
    const float v = ft[((size_t)s * Bsz + b) * HO + h * OUTo + o] * a;
    atomicAdd(&rst[((size_t)d * Bsz + b) * HO + h * OUTo + o], v);
}

// ---------------------------------------------------------------------------
extern "C" void kernel_launch(void* const* d_in, const int* in_sizes, int n_in,
                              void* d_out, int out_size, void* d_ws, size_t ws_size,
                              hipStream_t stream)
{
    const float* feat   = (const float*)d_in[0];
    const float* W1     = (const float*)d_in[1];
    const float* W2     = (const float*)d_in[2];
    const float* attn_l = (const float*)d_in[3];
    const int*   src    = (const int*)d_in[4];
    const int*   dst    = (const int*)d_in[5];
    const int*   taxi   = (const int*)d_in[6];

    const int N      = in_sizes[0] / (Bsz * INF);   // 20000
    const int E      = in_sizes[4];                 // 320000
    const int Mtotal = N * Bsz;                     // 40000 rows of ft
    const int NBH    = N * BH;
    const int EBH    = E * BH;
    const int EBHO   = EBH * OUTo;

    // workspace layout
    char* ws = (char*)d_ws;
    float*    ft   = (float*)ws;    ws += (size_t)Mtotal * HO * sizeof(float);  // 10.24 MB
    float*    el   = (float*)ws;    ws += (size_t)NBH * sizeof(float);
    unsigned* mkey = (unsigned*)ws; ws += (size_t)NBH * sizeof(unsigned);
    float*    ssum = (float*)ws;    ws += (size_t)NBH * sizeof(float);
    float*    ebuf = (float*)ws;    ws += (size_t)EBH * sizeof(float);          // 10.24 MB

    float* rst   = (float*)d_out;                   // [N,B,64]
    float* a_out = rst + (size_t)Mtotal * HO;       // [E,B,H]

    // zero accumulators (graph-capture-safe async memsets)
    hipMemsetAsync(rst,  0, (size_t)Mtotal * HO * sizeof(float), stream);
    hipMemsetAsync(mkey, 0, (size_t)NBH * sizeof(unsigned), stream);
    hipMemsetAsync(ssum, 0, (size_t)NBH * sizeof(float), stream);

    const int tiles = Mtotal / 16;                  // 2500 (Mtotal % 16 == 0)
    gemm_ft_kernel<<<(tiles + 3) / 4, 128, 0, stream>>>(feat, W1, W2, taxi, ft, Mtotal);
    el_kernel<<<(NBH + 255) / 256, 256, 0, stream>>>(ft, attn_l, el, NBH);
    edge_max_kernel<<<(EBH + 255) / 256, 256, 0, stream>>>(el, src, dst, ebuf, mkey, EBH);
    edge_exp_kernel<<<(EBH + 255) / 256, 256, 0, stream>>>(ebuf, mkey, dst, ssum, EBH);
    edge_norm_kernel<<<(EBH + 255) / 256, 256, 0, stream>>>(ebuf, ssum, dst, a_out, EBH);
    aggregate_kernel<<<(EBHO + 255) / 256, 256, 0, stream>>>(ft, a_out, src, dst, rst, EBHO);
}